// FretOnsetByStringFrameWiseLoss_35845797052655
// MI455X (gfx1250) — compile-verified
//
#include <hip/hip_runtime.h>
#include <hip/hip_bf16.h>
#include <stdint.h>

// Problem constants (from reference)
#define BB 64
#define SS 6
#define TT 2000
#define CC 25
#define STT (SS * TT)

typedef __attribute__((ext_vector_type(2))) float v2f;
typedef __attribute__((ext_vector_type(8))) float v8f;

// ---------------------------------------------------------------------------
// Full 32-lane (wave32) sum using one V_WMMA_F32_16X16X4_F32.
// A layout (16x4 f32): lanes 0-15 hold A[M=L][K=0..1] in vgpr0/1,
// lanes 16-31 hold A[M=L-16][K=2..3]. Put x in vgpr0, 0 in vgpr1, B = ones:
//   D[m][n] = x[lane m] + x[lane m+16]   (broadcast across all n)
// D layout: lane n<16 vgpr j = D[j][n]; lane n>=16 vgpr j = D[j+8][n-16].
// Sum the 8 D vgprs in-lane, then xor-16 combine -> full wave sum in all lanes.
// EXEC must be all ones at the call site (callers are convergence-safe).
// ---------------------------------------------------------------------------
__device__ __forceinline__ float wave_reduce_sum_f32(float x) {
#if __has_builtin(__builtin_amdgcn_wmma_f32_16x16x4_f32)
  v2f a;
  a[0] = x;
  a[1] = 0.0f;
  v2f b;
  b[0] = 1.0f;
  b[1] = 1.0f;
  v8f c = {};
  v8f d = __builtin_amdgcn_wmma_f32_16x16x4_f32(
      /*neg_a=*/false, a, /*neg_b=*/false, b,
      /*c_mod=*/(short)0, c, /*reuse_a=*/false, /*reuse_b=*/false);
  float r = ((d[0] + d[1]) + (d[2] + d[3])) + ((d[4] + d[5]) + (d[6] + d[7]));
  r += __shfl_xor(r, 16, 32);
  return r;
#else
  for (int m = 16; m >= 1; m >>= 1) x += __shfl_xor(x, m, 32);
  return x;
#endif
}

// ---------------------------------------------------------------------------
// Zero the 12-float accumulator workspace: ws[0..5]=fret sums, ws[6..11]=onset.
// ---------------------------------------------------------------------------
__global__ void zero_ws_kernel(float* __restrict__ ws) {
  if (threadIdx.x < 12) ws[threadIdx.x] = 0.0f;
}

// ---------------------------------------------------------------------------
// Fret CE kernel. Each block handles 256 contiguous rows of 25 logits.
// The 256*25 floats (25 KB, 16B-aligned since 256*25*4 % 16 == 0) are staged
// global->LDS with async b128 transfers (ASYNCcnt), then each thread reduces
// its own row out of LDS (stride-25 -> conflict-free across 64 banks).
// Per-string partials go through LDS ds_add_f32 buckets, then 6 global atomics.
// ---------------------------------------------------------------------------
#define FR_ROWS 256
#define FR_THREADS 256

__global__ void fret_loss_kernel(const float* __restrict__ xf,
                                 const int* __restrict__ tf,
                                 float* __restrict__ ws_fret) {
  __shared__ float tile[FR_ROWS * CC];  // 6400 floats = 25 KB
  __shared__ float sbuck[8];

  const int tid = threadIdx.x;
  if (tid < 8) sbuck[tid] = 0.0f;

  const long long R0 = (long long)blockIdx.x * FR_ROWS;
  const uint64_t gbase = (uint64_t)(uintptr_t)(xf + R0 * CC);
  const uint32_t lbase = (uint32_t)(uintptr_t)(&tile[0]);

  // 6400 x b128 transfers = 100 KB, 25 per thread, coalesced.
#pragma unroll
  for (int j = 0; j < CC; ++j) {
    const uint32_t off = (uint32_t)(j * FR_THREADS + tid) * 16u;
    const uint32_t laddr = lbase + off;
    const uint64_t gaddr = gbase + off;
    asm volatile("global_load_async_to_lds_b128 %0, %1, off"
                 :
                 : "v"(laddr), "v"(gaddr)
                 : "memory");
  }
  // Wait for this wave's async transfers; barrier makes all waves' LDS visible.
  asm volatile("s_wait_asynccnt 0" ::: "memory");
  __syncthreads();

  const float* row = &tile[tid * CC];
  float m = row[0];
#pragma unroll
  for (int c = 1; c < CC; ++c) m = fmaxf(m, row[c]);
  float se = 0.0f;
#pragma unroll
  for (int c = 0; c < CC; ++c) se += __expf(row[c] - m);

  const long long r = R0 + tid;
  const int tgt = tf[r];
  const float nll = (m + __logf(se)) - row[tgt];

  const int s = (int)((r / TT) % SS);
  atomicAdd(&sbuck[s], nll);  // ds_add_f32
  __syncthreads();
  if (tid < SS) atomicAdd(&ws_fret[tid], sbuck[tid]);
}

// ---------------------------------------------------------------------------
// Onset kernel: softmax over the batch axis (B=64) per (s,t) column.
// Block s handles string s; each wave owns 16 columns per chunk, lane L covers
// column (L&15) with batch half (L>>4): 32 x-values + 32 targets in registers.
// Column max / sums are combined across the two half-waves with xor-16
// shuffles; the per-wave running total is reduced with the WMMA reducer.
// Fully convergent (2000 = 8 waves * 16 cols * 125/8 chunks, no guards),
// so EXEC is all ones at the WMMA.
// ---------------------------------------------------------------------------
__global__ void onset_loss_kernel(const float* __restrict__ xo,
                                  const float* __restrict__ to,
                                  float* __restrict__ ws_on) {
  const int s = blockIdx.x;
  const int lane = threadIdx.x & 31;
  const int wave = threadIdx.x >> 5;
  const int half = lane >> 4;  // which 32 batch rows this lane covers
  const int cl = lane & 15;    // column within the wave's 16-column tile

  float acc = 0.0f;
  for (int t0 = wave * 16; t0 < TT; t0 += 16 * 8) {
    const int col = t0 + cl;        // always < TT (TT % 16 == 0)
    const int base = s * TT + col;  // [b][s][t] -> b*STT + base

    float xv[32], tv[32];
#pragma unroll
    for (int i = 0; i < 32; ++i) {
      const int idx = (half * 32 + i) * STT + base;
      xv[i] = xo[idx];
      tv[i] = to[idx];
    }

    float m = xv[0];
#pragma unroll
    for (int i = 1; i < 32; ++i) m = fmaxf(m, xv[i]);
    m = fmaxf(m, __shfl_xor(m, 16, 32));  // full-column max over B=64

    float se = 0.0f, st = 0.0f, sx = 0.0f;
#pragma unroll
    for (int i = 0; i < 32; ++i) {
      se += __expf(xv[i] - m);
      st += tv[i];
      sx += tv[i] * xv[i];
    }
    se += __shfl_xor(se, 16, 32);
    st += __shfl_xor(st, 16, 32);
    sx += __shfl_xor(sx, 16, 32);

    // -sum_b t[b]*(x[b]-LSE) = LSE*sum(t) - sum(t*x); LSE = m + log(se)
    const float on = (m + __logf(se)) * st - sx;
    acc += (half == 0) ? on : 0.0f;  // count each column once
  }

  const float tot = wave_reduce_sum_f32(acc);
  if (lane == 0) atomicAdd(&ws_on[s], tot);
}

// ---------------------------------------------------------------------------
// Finalize: ws[0..5] = sum_{b,t} nll  -> /B for batch mean; ws[6..11] = onset.
// d_out = [loss, fret_loss, onset_loss, fret_string[6], onset_string[6]].
// ---------------------------------------------------------------------------
__global__ void finalize_kernel(const float* __restrict__ ws,
                                float* __restrict__ out) {
  const int i = threadIdx.x;
  if (i < SS) {
    out[3 + i] = ws[i] * (1.0f / (float)BB);
    out[9 + i] = ws[SS + i];
  }
  if (i == 0) {
    float f = 0.0f, o = 0.0f;
    for (int k = 0; k < SS; ++k) {
      f += ws[k] * (1.0f / (float)BB);
      o += ws[SS + k];
    }
    out[1] = f;
    out[2] = o;
    out[0] = 0.5f * f + 0.5f * o;
  }
}

extern "C" void kernel_launch(void* const* d_in, const int* in_sizes, int n_in,
                              void* d_out, int out_size, void* d_ws,
                              size_t ws_size, hipStream_t stream) {
  const float* xf = (const float*)d_in[0];  // output_fret  [B,S,T,C] f32
  const int* tf = (const int*)d_in[1];      // target_fret  [B,S,T]   i32
  const float* xo = (const float*)d_in[2];  // output_onset [B,S,T]   f32
  const float* to = (const float*)d_in[3];  // target_onset [B,S,T]   f32
  float* out = (float*)d_out;
  float* ws = (float*)d_ws;  // 12 floats of accumulators

  zero_ws_kernel<<<1, 32, 0, stream>>>(ws);
  fret_loss_kernel<<<(BB * SS * TT) / FR_ROWS, FR_THREADS, 0, stream>>>(xf, tf, ws);
  onset_loss_kernel<<<SS, 256, 0, stream>>>(xo, to, ws + SS);
  finalize_kernel<<<1, 32, 0, stream>>>(ws, out);
}